// HGATConv_10969346474535
// MI455X (gfx1250) — compile-verified
//
#include <hip/hip_runtime.h>
#include <math.h>
#include <stdint.h>

typedef __attribute__((ext_vector_type(2))) float v2f;
typedef __attribute__((ext_vector_type(8))) float v8f;

#define D        128
#define DHEAD    64
#define MAXNORM  0.996f      // (1 - 4e-3)/sqrt(c), c = 1
#define MINNORM  1e-15f

// ---------- helpers ----------
__device__ __forceinline__ float wred32(float v) {
#pragma unroll
  for (int off = 16; off; off >>= 1) v += __shfl_xor(v, off, 32);
  return v;
}
// reduce within each 16-lane half (maps to the 2 attention heads)
__device__ __forceinline__ float wred16(float v) {
#pragma unroll
  for (int off = 8; off; off >>= 1) v += __shfl_xor(v, off, 32);
  return v;
}
__device__ __forceinline__ float artanh_f(float z) {
  z = fminf(fmaxf(z, -1.0f + 1e-7f), 1.0f - 1e-7f);
  return 0.5f * logf((1.0f + z) / (1.0f - z));
}
// order-preserving float->uint encoding for atomic segment-max
__device__ __forceinline__ unsigned encF(float f) {
  unsigned u = __float_as_uint(f);
  return (f >= 0.0f) ? (u | 0x80000000u) : ~u;
}
__device__ __forceinline__ float decF(unsigned u) {
  return (u & 0x80000000u) ? __uint_as_float(u & 0x7FFFFFFFu) : __uint_as_float(~u);
}

// ---------- 0: init scratch ----------
__global__ void k_init(float* __restrict__ agg, float* __restrict__ denom,
                       unsigned* __restrict__ amax, int N) {
  int i = blockIdx.x * blockDim.x + threadIdx.x;
  if (i < N * D) agg[i] = 0.0f;
  if (i < 2 * N) { denom[i] = 0.0f; amax[i] = 0x007FFFFFu; /* enc(-inf) */ }
}

// ---------- 1: w_hyp = proj(expmap0(weight)), b_hyp = proj(expmap0(bias)) ----------
__global__ void k_prep(const float* __restrict__ weight, const float* __restrict__ bias,
                       float* __restrict__ w_hyp, float* __restrict__ b_hyp) {
  int wid  = (blockIdx.x * blockDim.x + threadIdx.x) >> 5;
  int lane = threadIdx.x & 31;
  if (wid >= D + 1) return;                      // wave-uniform
  const float* src = (wid < D) ? (weight + (size_t)wid * D) : bias;
  float*       dst = (wid < D) ? (w_hyp  + (size_t)wid * D) : b_hyp;
  float4 u = ((const float4*)src)[lane];
  float s = wred32(u.x*u.x + u.y*u.y + u.z*u.z + u.w*u.w);
  float n = fmaxf(sqrtf(s), MINNORM);
  float sc = tanhf(n) / n;                       // expmap0, sqrt(c)=1
  u.x *= sc; u.y *= sc; u.z *= sc; u.w *= sc;
  s = wred32(u.x*u.x + u.y*u.y + u.z*u.z + u.w*u.w);
  n = fmaxf(sqrtf(s), MINNORM);
  if (n > MAXNORM) { float f = MAXNORM / n; u.x*=f; u.y*=f; u.z*=f; u.w*=f; }
  ((float4*)dst)[lane] = u;
}

// ---------- 2: mx = x @ w_hyp^T via V_WMMA_F32_16X16X4_F32 ----------
// Weights staged in LDS with gfx1250 async global->LDS DMA (ASYNCcnt-tracked).
__global__ void k_gemm(const float* __restrict__ x, const float* __restrict__ w_hyp,
                       float* __restrict__ mx, int N) {
  __shared__ float wlds[D * D];                  // 64 KB of the 320 KB WGP LDS

  // async copy: 4096 x B128 transfers, 16 per thread, straight into LDS
  {
    for (int i = threadIdx.x; i < (D * D) / 4; i += blockDim.x) {
      unsigned ldsoff = (unsigned)(uintptr_t)(&wlds[0]) + (unsigned)(i * 16);
      unsigned long long ga =
          (unsigned long long)(uintptr_t)((const char*)w_hyp + (size_t)i * 16);
      asm volatile("global_load_async_to_lds_b128 %0, %1, off"
                   :: "v"(ldsoff), "v"(ga) : "memory");
    }
    asm volatile("s_wait_asynccnt 0" ::: "memory");
  }
  __syncthreads();

  const int wpb   = blockDim.x >> 5;
  const int wid   = blockIdx.x * wpb + (threadIdx.x >> 5);
  const int nw    = gridDim.x * wpb;
  const int lane  = threadIdx.x & 31;
  const int half  = lane >> 4;                   // wave32 half select
  const int nm    = lane & 15;                   // M (A) / N (B,C,D) index
  const int tiles = (N + 15) >> 4;

  for (int tile = wid; tile < tiles; tile += nw) {   // wave-uniform: EXEC all 1s
    const int i0 = tile << 4;
    v8f acc[8];
#pragma unroll
    for (int t = 0; t < 8; ++t)
#pragma unroll
      for (int r = 0; r < 8; ++r) acc[t][r] = 0.0f;

    int arow = i0 + nm; if (arow >= N) arow = N - 1;     // clamp (store guarded)
    const float* xrow = x + (size_t)arow * D + half * 2; // A: K∈{0,1}|{2,3} per half

#pragma unroll 4
    for (int k = 0; k < D; k += 4) {
      v2f a; a.x = xrow[k]; a.y = xrow[k + 1];
#pragma unroll
      for (int t = 0; t < 8; ++t) {
        const float* wp = &wlds[(t * 16 + nm) * D + k + half * 2]; // B[k'][n]=w[n][k']
        v2f b; b.x = wp[0]; b.y = wp[1];
        acc[t] = __builtin_amdgcn_wmma_f32_16x16x4_f32(
            false, a, false, b, (short)0, acc[t], false, false);
      }
    }
    // C/D layout: lane holds (M = r + 8*half, N = nm) in VGPR r
#pragma unroll
    for (int t = 0; t < 8; ++t)
#pragma unroll
      for (int r = 0; r < 8; ++r) {
        int row = i0 + r + half * 8;
        if (row < N) mx[(size_t)row * D + t * 16 + nm] = acc[t][r];
      }
  }
}

// ---------- 3: per-row mobius_matvec tail + proj + mobius_add(bias) + proj + logmap0 ----------
__global__ void k_row(const float* __restrict__ x, float* __restrict__ hio,
                      const float* __restrict__ b_hyp, int N) {
  int wid  = blockIdx.x * (blockDim.x >> 5) + (threadIdx.x >> 5);
  int nw   = gridDim.x * (blockDim.x >> 5);
  int lane = threadIdx.x & 31;
  float4 bv = ((const float4*)b_hyp)[lane];
  float y2 = wred32(bv.x*bv.x + bv.y*bv.y + bv.z*bv.z + bv.w*bv.w);

  for (int row = wid; row < N; row += nw) {
    float4 xv = ((const float4*)(x   + (size_t)row * D))[lane];
    float4 mv = ((const float4*)(hio + (size_t)row * D))[lane];
    float sx = wred32(xv.x*xv.x + xv.y*xv.y + xv.z*xv.z + xv.w*xv.w);
    float sm = wred32(mv.x*mv.x + mv.y*mv.y + mv.z*mv.z + mv.w*mv.w);
    float xn  = fmaxf(sqrtf(sx), MINNORM);
    float mxn = fmaxf(sqrtf(sm), MINNORM);
    float scale = tanhf(mxn / xn * artanh_f(xn)) / mxn;   // mobius_matvec
    if (sm == 0.0f) scale = 0.0f;                          // all-zero mx row
    float h0 = mv.x*scale, h1 = mv.y*scale, h2 = mv.z*scale, h3 = mv.w*scale;
    // proj
    float s = wred32(h0*h0 + h1*h1 + h2*h2 + h3*h3);
    float n = fmaxf(sqrtf(s), MINNORM);
    if (n > MAXNORM) { float f = MAXNORM/n; h0*=f; h1*=f; h2*=f; h3*=f; }
    // mobius_add(h, b_hyp), c = 1
    float x2 = wred32(h0*h0 + h1*h1 + h2*h2 + h3*h3);
    float xy = wred32(h0*bv.x + h1*bv.y + h2*bv.z + h3*bv.w);
    float cnum1 = 1.0f + 2.0f*xy + y2;
    float cnum2 = 1.0f - x2;
    float den = fmaxf(1.0f + 2.0f*xy + x2*y2, MINNORM);
    h0 = (cnum1*h0 + cnum2*bv.x) / den;  h1 = (cnum1*h1 + cnum2*bv.y) / den;
    h2 = (cnum1*h2 + cnum2*bv.z) / den;  h3 = (cnum1*h3 + cnum2*bv.w) / den;
    // proj
    s = wred32(h0*h0 + h1*h1 + h2*h2 + h3*h3);
    n = fmaxf(sqrtf(s), MINNORM);
    if (n > MAXNORM) { float f = MAXNORM/n; h0*=f; h1*=f; h2*=f; h3*=f; }
    // logmap0
    s = wred32(h0*h0 + h1*h1 + h2*h2 + h3*h3);
    n = fmaxf(sqrtf(s), MINNORM);
    float l = artanh_f(n) / n;
    float4 o; o.x = h0*l; o.y = h1*l; o.z = h2*l; o.w = h3*l;
    ((float4*)(hio + (size_t)row * D))[lane] = o;          // in-place -> h_t
  }
}

// ---------- 4: per-edge attention logits + leaky_relu + segment max ----------
__global__ void k_alpha(const float* __restrict__ ht, const int* __restrict__ e0,
                        const int* __restrict__ e1, const float* __restrict__ att_i,
                        const float* __restrict__ att_j, float* __restrict__ alpha,
                        unsigned* __restrict__ amax, int E, int N) {
  int wid  = blockIdx.x * (blockDim.x >> 5) + (threadIdx.x >> 5);
  int nw   = gridDim.x * (blockDim.x >> 5);
  int lane = threadIdx.x & 31;
  float4 ai = ((const float4*)att_i)[lane];   // cols 4*lane..+3, head = lane>>4
  float4 aj = ((const float4*)att_j)[lane];
  int EN = E + N;
  for (int e = wid; e < EN; e += nw) {
    int ei = (e < E) ? e0[e] : (e - E);
    int ej = (e < E) ? e1[e] : (e - E);
    float4 xi = ((const float4*)(ht + (size_t)ei * D))[lane];
    float4 xj = ((const float4*)(ht + (size_t)ej * D))[lane];
    float p = xi.x*ai.x + xi.y*ai.y + xi.z*ai.z + xi.w*ai.w
            + xj.x*aj.x + xj.y*aj.y + xj.z*aj.z + xj.w*aj.w;
    p = wred16(p);                             // per-head partial sums
    if ((lane & 15) == 0) {
      float a = (p > 0.0f) ? p : 0.2f * p;     // leaky_relu 0.2
      int h = lane >> 4;
      alpha[(size_t)e * 2 + h] = a;
      atomicMax(&amax[(size_t)ei * 2 + h], encF(a));
    }
  }
}

// ---------- 5: alpha <- exp(alpha - max); denom segment sum ----------
__global__ void k_expdenom(const int* __restrict__ e0, float* __restrict__ alpha,
                           const unsigned* __restrict__ amax, float* __restrict__ denom,
                           int E, int N) {
  int t = blockIdx.x * blockDim.x + threadIdx.x;
  int EN = E + N;
  if (t >= EN) return;
  int ei = (t < E) ? e0[t] : (t - E);
  float a0 = alpha[(size_t)t*2],     a1 = alpha[(size_t)t*2 + 1];
  float m0 = decF(amax[(size_t)ei*2]), m1 = decF(amax[(size_t)ei*2 + 1]);
  float v0 = expf(a0 - m0), v1 = expf(a1 - m1);
  alpha[(size_t)t*2] = v0; alpha[(size_t)t*2 + 1] = v1;
  atomicAdd(&denom[(size_t)ei*2],     v0);
  atomicAdd(&denom[(size_t)ei*2 + 1], v1);
}

// ---------- 6: weighted aggregation (segment sum of x_j * alpha) ----------
__global__ void k_agg(const float* __restrict__ ht, const int* __restrict__ e0,
                      const int* __restrict__ e1, const float* __restrict__ alpha,
                      const float* __restrict__ denom, float* __restrict__ agg,
                      int E, int N) {
  int wid  = blockIdx.x * (blockDim.x >> 5) + (threadIdx.x >> 5);
  int nw   = gridDim.x * (blockDim.x >> 5);
  int lane = threadIdx.x & 31;
  int EN = E + N;
  for (int e = wid; e < EN; e += nw) {
    int ei = (e < E) ? e0[e] : (e - E);
    int ej = (e < E) ? e1[e] : (e - E);
    float w0 = alpha[(size_t)e*2]     / fmaxf(denom[(size_t)ei*2],     1e-16f);
    float w1 = alpha[(size_t)e*2 + 1] / fmaxf(denom[(size_t)ei*2 + 1], 1e-16f);
    float w = (lane < 16) ? w0 : w1;           // head = lane>>4
    float4 xj = ((const float4*)(ht + (size_t)ej * D))[lane];
    float* dst = agg + (size_t)ei * D + 4 * lane;
    atomicAdd(dst + 0, xj.x * w);
    atomicAdd(dst + 1, xj.y * w);
    atomicAdd(dst + 2, xj.z * w);
    atomicAdd(dst + 3, xj.w * w);
  }
}

// ---------- 7: mean heads -> expmap0 -> proj -> logmap0 -> leaky -> expmap0 -> proj ----------
__global__ void k_final(const float* __restrict__ agg, float* __restrict__ out, int N) {
  int wid  = blockIdx.x * (blockDim.x >> 5) + (threadIdx.x >> 5);
  int nw   = gridDim.x * (blockDim.x >> 5);
  int lane = threadIdx.x & 31;
  int d0 = 2 * lane, d1 = 2 * lane + 1;
  for (int n = wid; n < N; n += nw) {
    const float* a = agg + (size_t)n * D;
    float g0 = 0.5f * (a[d0] + a[DHEAD + d0]);
    float g1 = 0.5f * (a[d1] + a[DHEAD + d1]);
    float s  = wred32(g0*g0 + g1*g1);
    float nn = fmaxf(sqrtf(s), MINNORM);
    float sc = tanhf(nn) / nn;  g0 *= sc; g1 *= sc;          // expmap0
    s = wred32(g0*g0 + g1*g1);  nn = fmaxf(sqrtf(s), MINNORM);
    if (nn > MAXNORM) { float f = MAXNORM/nn; g0*=f; g1*=f; } // proj
    s = wred32(g0*g0 + g1*g1);  nn = fmaxf(sqrtf(s), MINNORM);
    float l = artanh_f(nn) / nn; g0 *= l; g1 *= l;           // logmap0
    g0 = (g0 > 0.0f) ? g0 : 0.01f * g0;                      // leaky 0.01
    g1 = (g1 > 0.0f) ? g1 : 0.01f * g1;
    s = wred32(g0*g0 + g1*g1);  nn = fmaxf(sqrtf(s), MINNORM);
    sc = tanhf(nn) / nn; g0 *= sc; g1 *= sc;                 // expmap0 (c_out)
    s = wred32(g0*g0 + g1*g1);  nn = fmaxf(sqrtf(s), MINNORM);
    if (nn > MAXNORM) { float f = MAXNORM/nn; g0*=f; g1*=f; } // proj
    out[(size_t)n * DHEAD + d0] = g0;
    out[(size_t)n * DHEAD + d1] = g1;
  }
}

extern "C" void kernel_launch(void* const* d_in, const int* in_sizes, int n_in,
                              void* d_out, int out_size, void* d_ws, size_t ws_size,
                              hipStream_t stream) {
  const float* x      = (const float*)d_in[0];
  const int*   eidx   = (const int*)  d_in[1];
  const float* weight = (const float*)d_in[2];
  const float* bias   = (const float*)d_in[3];
  const float* att_i  = (const float*)d_in[4];
  const float* att_j  = (const float*)d_in[5];
  const int N  = in_sizes[0] / D;
  const int E  = in_sizes[1] / 2;
  const int EN = E + N;
  const int* e0 = eidx;
  const int* e1 = eidx + E;

  char* ws = (char*)d_ws;
  size_t off = 0;
  auto take = [&](size_t bytes) -> void* {
    void* p = ws + off;
    off = (off + bytes + 255) & ~(size_t)255;
    return p;
  };
  float*    w_hyp = (float*)   take((size_t)D * D * sizeof(float));
  float*    b_hyp = (float*)   take((size_t)D * sizeof(float));
  float*    ht    = (float*)   take((size_t)N * D * sizeof(float));   // mx -> h_t
  float*    alpha = (float*)   take((size_t)EN * 2 * sizeof(float));
  unsigned* amax  = (unsigned*)take((size_t)N * 2 * sizeof(unsigned));
  float*    denom = (float*)   take((size_t)N * 2 * sizeof(float));
  float*    agg   = (float*)   take((size_t)N * D * sizeof(float));
  (void)ws_size; (void)n_in; (void)out_size;

  const int tiles = (N + 15) >> 4;
  k_init    <<<(N * D + 255) / 256, 256, 0, stream>>>(agg, denom, amax, N);
  k_prep    <<<((D + 1) * 32 + 255) / 256, 256, 0, stream>>>(weight, bias, w_hyp, b_hyp);
  k_gemm    <<<(tiles + 7) / 8, 256, 0, stream>>>(x, w_hyp, ht, N);
  k_row     <<<(N + 7) / 8, 256, 0, stream>>>(x, ht, b_hyp, N);
  k_alpha   <<<(EN + 7) / 8, 256, 0, stream>>>(ht, e0, e1, att_i, att_j, alpha, amax, E, N);
  k_expdenom<<<(EN + 255) / 256, 256, 0, stream>>>(e0, alpha, amax, denom, E, N);
  k_agg     <<<(EN + 7) / 8, 256, 0, stream>>>(ht, e0, e1, alpha, denom, agg, E, N);
  k_final   <<<(N + 7) / 8, 256, 0, stream>>>(agg, (float*)d_out, N);
}